// VectorQuantizer_72636486910348
// MI455X (gfx1250) — compile-verified
//
#include <hip/hip_runtime.h>
#include <cmath>

typedef float v2f __attribute__((ext_vector_type(2)));
typedef float v8f __attribute__((ext_vector_type(8)));
typedef unsigned int u32x4 __attribute__((ext_vector_type(4)));
typedef int i32x4 __attribute__((ext_vector_type(4)));
typedef int i32x8 __attribute__((ext_vector_type(8)));

#define K_CODES 512
#define DIM     64
#define NROWS   (32 * 64 * 64)   // 131072 rows (b*h*w), hw-block = 4096
#define BETA    0.25f

#define STAGE_CODES 128          // codes staged in LDS per step (4 stages of 128)
#define LDS_STRIDE  68           // 64 data dwords + 4 pad dwords -> conflict-free B reads

// ---------------- kernel 0: init (emb half-norms, zero counts) ----------------
__global__ __launch_bounds__(512) void vq_init(const float* __restrict__ emb,
                                               float* __restrict__ halfNorm,
                                               int* __restrict__ counts) {
  int k = blockIdx.x * blockDim.x + threadIdx.x;
  if (k < K_CODES) {
    const float* e = emb + k * DIM;
    float s = 0.f;
    #pragma unroll
    for (int d = 0; d < DIM; ++d) s += e[d] * e[d];
    halfNorm[k] = 0.5f * s;
    counts[k] = 0;
  }
}

// ---------------- kernel 1: TDM-staged WMMA score + fused argmax ---------------
// One wave owns a 16-row tile; the block stages 128 codes at a time into LDS
// via the Tensor Data Mover (with padding for bank-conflict-free ds_load_b64),
// then each wave runs 8 code-tiles of fp32 WMMA (16x16x4) against the stage.
__global__ __launch_bounds__(256) void vq_argmin_wmma(const float* __restrict__ x,
                                                      const float* __restrict__ emb,
                                                      const float* __restrict__ halfNorm,
                                                      int* __restrict__ idxOut) {
  __shared__ float s_emb[STAGE_CODES * LDS_STRIDE];   // 34,816 B

  const int lane    = threadIdx.x & 31;
  const int waveId  = blockIdx.x * (blockDim.x >> 5) + (threadIdx.x >> 5);
  const int rowBase = waveId * 16;
  const int rlane   = lane & 15;
  const int khi     = (lane >> 4) * 2;        // K sub-offset: lanes 0-15 -> {0,1}, 16-31 -> {2,3}
  const int n       = rowBase + rlane;        // logical row (same for lane and lane+16)

  // x is (b, d, h, w): x_flat[n, dd] = x[(b*DIM + dd)*4096 + hw], n = b*4096 + hw
  const float* xr = x + ((size_t)(n >> 12) * (DIM * 4096)) + (n & 4095);

  // Preload A matrix (16 rows x 64 dims) into registers: 2 VGPRs per K=4 step.
  float a0[16], a1[16];
  #pragma unroll
  for (int s = 0; s < 16; ++s) {
    int dd = s * 4 + khi;
    a0[s] = xr[(size_t)dd * 4096];
    a1[s] = xr[(size_t)(dd + 1) * 4096];
  }

  float best[8];
  int   bidx[8];
  #pragma unroll
  for (int r = 0; r < 8; ++r) { best[r] = -3.4e38f; bidx[r] = 0; }

  for (int stage = 0; stage < K_CODES / STAGE_CODES; ++stage) {
#if __has_builtin(__builtin_amdgcn_tensor_load_to_lds)
    if (threadIdx.x < 32) {
      // ---- Tensor DMA descriptor (D#): 64x128 f32 tile, global -> LDS ----
      // Group0: count=1 | lds_addr | global_addr | type=2
      unsigned long long gaddr =
          (unsigned long long)(uintptr_t)(emb + (size_t)stage * STAGE_CODES * DIM);
      unsigned ldsOff = (unsigned)(uintptr_t)&s_emb[0];  // low 32b of LDS aperture = byte offset
      u32x4 g0;
      g0[0] = 1u;                                   // count=1, is_restore=0, gather=0
      g0[1] = ldsOff;                               // lds_addr (bytes)
      g0[2] = (unsigned)(gaddr & 0xFFFFFFFFu);      // global_addr[31:0]
      g0[3] = (unsigned)((gaddr >> 32) & 0x01FFFFFFu) | 0x80000000u;  // addr[56:32] | type=2
      // Group1: data_size=4B, pad 4 dwords after every 64 dwords,
      //         tensor 64 x 128 (stride 64), tile 64 x 128
      i32x8 g1;
      g1[0] = (int)((2u << 16) | (1u << 20) | (5u << 22) | (3u << 25)); // dsize|pad_en|int=64dw|amt=4dw
      g1[1] = (int)(64u << 16);                     // tensor_dim0 = 64
      g1[2] = (int)(128u << 16);                    // tensor_dim1 = 128
      g1[3] = (int)(64u << 16);                     // tile_dim0 = 64
      g1[4] = 128;                                  // tile_dim1 = 128, tile_dim2 = 0
      g1[5] = 64;                                   // tensor_dim0_stride = 64
      g1[6] = 0;
      g1[7] = 0;
      i32x4 gz = {0, 0, 0, 0};
#if __clang_major__ >= 23
      i32x8 gz8 = {0, 0, 0, 0, 0, 0, 0, 0};
      __builtin_amdgcn_tensor_load_to_lds(g0, g1, gz, gz, gz8, 0);
#else
      __builtin_amdgcn_tensor_load_to_lds(g0, g1, gz, gz, 0);
#endif
      __builtin_amdgcn_s_wait_tensorcnt(0);
    }
#else
    // Fallback: cooperative staging with explicit padding
    for (int i = threadIdx.x; i < STAGE_CODES * DIM; i += 256) {
      int c = i >> 6, d = i & 63;
      s_emb[c * LDS_STRIDE + d] = emb[((size_t)stage * STAGE_CODES + c) * DIM + d];
    }
#endif
    __syncthreads();

    for (int tt = 0; tt < STAGE_CODES / 16; ++tt) { // 8 code tiles per stage
      const int codeLocal = tt * 16 + rlane;        // this lane's B column within stage
      const int code      = stage * STAGE_CODES + codeLocal;
      const float* ec     = s_emb + codeLocal * LDS_STRIDE + khi; // conflict-free ds_load_b64
      v8f c = {};
      #pragma unroll
      for (int s = 0; s < 16; ++s) {                // accumulate D=64 in K=4 chunks
        v2f A; A.x = a0[s];     A.y = a1[s];
        v2f B; B.x = ec[s * 4]; B.y = ec[s * 4 + 1];
        c = __builtin_amdgcn_wmma_f32_16x16x4_f32(false, A, false, B,
                                                  (short)0, c, false, false);
      }
      const float hn = halfNorm[code];
      // C layout: VGPR r holds M = r (lanes 0-15) or M = r+8 (lanes 16-31), N = lane%16
      #pragma unroll
      for (int r = 0; r < 8; ++r) {
        float sc = c[r] - hn;
        if (sc > best[r]) { best[r] = sc; bidx[r] = code; }
      }
    }
    __syncthreads();   // stage consumed before next TDM overwrites LDS
  }

  // Argmax across the 16 lanes of each half-wave (codes live on the N/lane axis).
  #pragma unroll
  for (int m = 1; m < 16; m <<= 1) {
    #pragma unroll
    for (int r = 0; r < 8; ++r) {
      float os = __shfl_xor(best[r], m, 32);
      int   oi = __shfl_xor(bidx[r], m, 32);
      if (os > best[r] || (os == best[r] && oi < bidx[r])) { best[r] = os; bidx[r] = oi; }
    }
  }
  if (rlane == 0) {
    const int mBase = rowBase + (lane >> 4) * 8;   // lanes 0..15 -> rows 0-7, 16..31 -> rows 8-15
    #pragma unroll
    for (int r = 0; r < 8; ++r) idxOut[mBase + r] = bidx[r];
  }
}

// ---------------- kernel 2: emit xq, one_hot, SSE partials, counts -------------
__global__ __launch_bounds__(256) void vq_emit(const float* __restrict__ x,
                                               const float* __restrict__ emb,
                                               const int* __restrict__ idx,
                                               float* __restrict__ xq,
                                               float* __restrict__ oneHot,
                                               int* __restrict__ counts,
                                               float* __restrict__ partialSSE) {
  __shared__ int   s_idx[256];
  __shared__ float s_red[256];
  const int tid = threadIdx.x;
  const int n   = blockIdx.x * 256 + tid;
  const int code = idx[n];
  s_idx[tid] = code;
  atomicAdd(&counts[code], 1);

  // xq output is (b, d, h, w): same strided layout as x.
  const size_t xbase = ((size_t)(n >> 12) * (DIM * 4096)) + (n & 4095);
  const float* e = emb + code * DIM;
  float sse = 0.f;
  #pragma unroll
  for (int dd = 0; dd < DIM; ++dd) {          // coalesced across tid for each dd
    float ev  = e[dd];
    float xv  = x[xbase + (size_t)dd * 4096];
    float dlt = ev - xv;
    sse += dlt * dlt;
    xq[xbase + (size_t)dd * 4096] = ev;
  }
  s_red[tid] = sse;
  __syncthreads();
  for (int s = 128; s > 0; s >>= 1) {         // deterministic tree reduction
    if (tid < s) s_red[tid] += s_red[tid + s];
    __syncthreads();
  }
  if (tid == 0) partialSSE[blockIdx.x] = s_red[0];

  // one_hot: 256 rows x 512 floats, float4 stores (the 256 MB stream).
  float4* oh = (float4*)(oneHot + (size_t)blockIdx.x * 256 * K_CODES);
  for (int i = tid; i < 256 * 128; i += 256) {
    const int row  = i >> 7;
    const int col4 = (i & 127) * 4;
    const int ci   = s_idx[row];
    float4 v;
    v.x = (col4     == ci) ? 1.f : 0.f;
    v.y = (col4 + 1 == ci) ? 1.f : 0.f;
    v.z = (col4 + 2 == ci) ? 1.f : 0.f;
    v.w = (col4 + 3 == ci) ? 1.f : 0.f;
    oh[(size_t)row * 128 + (i & 127)] = v;
  }
}

// ---------------- kernel 3: finalize scalars ----------------------------------
__global__ __launch_bounds__(512) void vq_finalize(const float* __restrict__ partialSSE,
                                                   const int* __restrict__ counts,
                                                   float* __restrict__ out) {
  __shared__ float s1[512];
  __shared__ float s2[512];
  const int tid = threadIdx.x;
  s1[tid] = partialSSE[tid];
  float p = (float)counts[tid] / (float)NROWS;
  s2[tid] = p * logf(p + 1e-10f);
  __syncthreads();
  for (int s = 256; s > 0; s >>= 1) {
    if (tid < s) { s1[tid] += s1[tid + s]; s2[tid] += s2[tid + s]; }
    __syncthreads();
  }
  if (tid == 0) {
    float mse = s1[0] / (float)((size_t)NROWS * DIM);
    out[0] = (1.0f + BETA) * mse;   // vq_loss (codebook + beta*commitment; equal fwd)
    out[1] = mse;                   // codebook_loss
    out[2] = mse;                   // commitment_loss
    out[3 + 8388608] = expf(-s2[0]);// perplexity
  }
}

extern "C" void kernel_launch(void* const* d_in, const int* in_sizes, int n_in,
                              void* d_out, int out_size, void* d_ws, size_t ws_size,
                              hipStream_t stream) {
  const float* x   = (const float*)d_in[0];   // (32, 64, 64, 64) f32
  const float* emb = (const float*)d_in[1];   // (512, 64) f32

  float* out    = (float*)d_out;
  float* xq     = out + 3;                    // 8388608 floats
  float* oneHot = out + 3 + 8388608 + 1;      // 67108864 floats (16B-aligned offset)

  char*  ws       = (char*)d_ws;
  float* halfNorm = (float*)(ws);                       // 512 f32
  int*   counts   = (int*)  (ws + 2048);                // 512 i32
  int*   idx      = (int*)  (ws + 4096);                // 131072 i32
  float* partial  = (float*)(ws + 4096 + NROWS * 4);    // 512 f32 (one per emit block)

  vq_init       <<<1, 512, 0, stream>>>(emb, halfNorm, counts);
  vq_argmin_wmma<<<NROWS / 16 / 8, 256, 0, stream>>>(x, emb, halfNorm, idx); // 1024 blocks, 8 waves each
  vq_emit       <<<NROWS / 256, 256, 0, stream>>>(x, emb, idx, xq, oneHot, counts, partial);
  vq_finalize   <<<1, 512, 0, stream>>>(partial, counts, out);
}